// PG_LOSS_28853590294681
// MI455X (gfx1250) — compile-verified
//
#include <hip/hip_runtime.h>
#include <hip/hip_bf16.h>

// Problem dimensions (fixed by the reference)
#define STATE_DIM 32
#define ACTION_DIM 16
#define T_LEN 512
#define S_LEN 2048
#define TSZ ((size_t)T_LEN * (size_t)S_LEN)   // plane stride for [*, T, S] tensors
#define NCHUNK 16
#define TCHUNK (T_LEN / NCHUNK)               // 32 timesteps per chunk
#define LOG2PI_F 1.8378770664093453f

typedef __attribute__((ext_vector_type(2))) float v2f;
typedef __attribute__((ext_vector_type(8))) float v8f;

// ---------------- workspace layout (bytes) ----------------
// [0..255]                      : loss accumulator (float at offset 0)
// [256 .. 256+128K)             : chunk_sums [NCHUNK][S]
// [256+128K .. 256+256K)        : tail       [NCHUNK][S]
// [256+256K .. +2K)             : rowsum     [T]
#define WS_LOSS_OFF    0
#define WS_CSUM_OFF    256
#define WS_TAIL_OFF    (256 + NCHUNK * S_LEN * 4)
#define WS_ROWSUM_OFF  (256 + 2 * NCHUNK * S_LEN * 4)

__global__ void pg_init(float* lossAcc) {
    if (threadIdx.x == 0 && blockIdx.x == 0) lossAcc[0] = 0.0f;
}

// chunk_sums[c][s] = sum_{t in chunk c} rewards[t][s]
__global__ __launch_bounds__(256) void pg_chunk_sum(const float* __restrict__ rewards,
                                                    float* __restrict__ csum) {
    int idx = blockIdx.x * blockDim.x + threadIdx.x;   // [0, NCHUNK*S)
    int c = idx / S_LEN;
    int s = idx - c * S_LEN;
    float acc = 0.0f;
    int t0 = c * TCHUNK;
    #pragma unroll 4
    for (int t = t0; t < t0 + TCHUNK; ++t)
        acc += rewards[(size_t)t * S_LEN + s];
    csum[idx] = acc;
}

// tail[c][s] = sum over chunks strictly after c
__global__ __launch_bounds__(256) void pg_tail(const float* __restrict__ csum,
                                               float* __restrict__ tail) {
    int s = blockIdx.x * blockDim.x + threadIdx.x;     // [0, S)
    float run = 0.0f;
    #pragma unroll
    for (int c = NCHUNK - 1; c >= 0; --c) {
        tail[c * S_LEN + s] = run;
        run += csum[c * S_LEN + s];
    }
}

// rowsum[t] = sum_s rewards[t][s]
__global__ __launch_bounds__(256) void pg_row_sum(const float* __restrict__ rewards,
                                                  float* __restrict__ rowsum) {
    int t = blockIdx.x;
    float a = 0.0f;
    for (int s = threadIdx.x; s < S_LEN; s += 256)
        a += rewards[(size_t)t * S_LEN + s];
    __shared__ float red[256];
    red[threadIdx.x] = a;
    __syncthreads();
    for (int off = 128; off > 0; off >>= 1) {
        if (threadIdx.x < off) red[threadIdx.x] += red[threadIdx.x + off];
        __syncthreads();
    }
    if (threadIdx.x == 0) rowsum[t] = red[0];
}

// expected_reward[t] = (1/S) * sum_{t'>=t} rowsum[t']   (suffix scan, tiny)
__global__ void pg_expected(const float* __restrict__ rowsum, float* __restrict__ outT) {
    if (threadIdx.x == 0 && blockIdx.x == 0) {
        float run = 0.0f;
        for (int t = T_LEN - 1; t >= 0; --t) {
            run += rowsum[t];
            outT[t] = run * (1.0f / (float)S_LEN);
        }
    }
}

// Fused: fp32 WMMA GEMM (mean = W @ states) + Gaussian loglik + reward-to-go
// weighting. One wave owns 16 s-columns and one t-chunk (walked backwards).
__global__ __launch_bounds__(256) void pg_main(const float* __restrict__ states,
                                               const float* __restrict__ actions,
                                               const float* __restrict__ rewards,
                                               const float* __restrict__ W,
                                               const float* __restrict__ log_std,
                                               const float* __restrict__ tail,
                                               float* __restrict__ lossAcc) {
    const int lane  = threadIdx.x & 31;
    const int wave  = threadIdx.x >> 5;
    const int laneN = lane & 15;          // column / M-row index within half
    const int hi    = lane >> 4;          // 0 = low lane half, 1 = high
    const int kOff  = hi * 2;             // A/B fragment K offset for this half
    const int aOff  = hi * 8;             // C/D fragment M offset for this half

    const int s0     = blockIdx.x * 128 + wave * 16;   // 8 waves * 16 cols
    const int chunk  = blockIdx.y;
    const int tStart = chunk * TCHUNK;
    const int tEnd   = tStart + TCHUNK;
    const int sCol   = s0 + laneN;

    // A fragments: W is the 16x32 A-matrix, resident for the whole kernel.
    // Layout (ISA 32-bit A 16x4): lane%16 = M, v0/v1 hold K = kOff / kOff+1.
    v2f aFrag[8];
    #pragma unroll
    for (int k = 0; k < 8; ++k) {
        aFrag[k].x = W[laneN * STATE_DIM + 4 * k + kOff];
        aFrag[k].y = W[laneN * STATE_DIM + 4 * k + kOff + 1];
    }

    float invStd[8];
    #pragma unroll
    for (int r = 0; r < 8; ++r) invStd[r] = __expf(-log_std[aOff + r]);
    float sumLog = 0.0f;
    #pragma unroll
    for (int a = 0; a < ACTION_DIM; ++a) sumLog += log_std[a];
    const float Kconst = sumLog + 0.5f * (float)ACTION_DIM * LOG2PI_F;

    float running  = tail[chunk * S_LEN + sCol];  // reward-to-go past this chunk
    float acc_loss = 0.0f;

    for (int t = tEnd - 1; t >= tStart; --t) {
        const size_t colBase = (size_t)t * S_LEN + sCol;

        running += rewards[colBase];              // cumulative[t][sCol]

        // mean tile = W(16x32) x states(32x16) via 8 chained fp32 WMMAs
        v8f acc = {};
        #pragma unroll
        for (int k = 0; k < 8; ++k) {
            v2f b;
            b.x = states[(size_t)(4 * k + kOff)     * TSZ + colBase];
            b.y = states[(size_t)(4 * k + kOff + 1) * TSZ + colBase];
#if defined(__gfx1250__)
            acc = __builtin_amdgcn_wmma_f32_16x16x4_f32(
                /*neg_a=*/false, aFrag[k], /*neg_b=*/false, b,
                /*c_mod=*/(short)0, acc, /*reuse_a=*/false, /*reuse_b=*/false);
#endif
        }

        // z = (actions - mean) * exp(-log_std); partial sum of z^2 per lane-half
        float part = 0.0f;
        #pragma unroll
        for (int r = 0; r < 8; ++r) {
            float act = actions[(size_t)(aOff + r) * TSZ + colBase];
            float z = (act - acc[r]) * invStd[r];
            part = fmaf(z, z, part);
        }
        // combine the two lane halves (M=0..7 and M=8..15 of this column)
        part += __shfl_xor(part, 16, 32);

        float loglik = -0.5f * part - Kconst;
        acc_loss += running * loglik;   // duplicated in both halves; /2 at end
    }

    __shared__ float red[256];
    red[threadIdx.x] = acc_loss;
    __syncthreads();
    for (int off = 128; off > 0; off >>= 1) {
        if (threadIdx.x < off) red[threadIdx.x] += red[threadIdx.x + off];
        __syncthreads();
    }
    if (threadIdx.x == 0) atomicAdd(lossAcc, red[0]);
}

__global__ void pg_finalize(const float* __restrict__ lossAcc, float* __restrict__ out0) {
    if (threadIdx.x == 0 && blockIdx.x == 0) {
        // x0.5 undoes lane-half duplication; negate + normalize by S*T
        out0[0] = -(lossAcc[0] * 0.5f) / ((float)S_LEN * (float)T_LEN);
    }
}

extern "C" void kernel_launch(void* const* d_in, const int* in_sizes, int n_in,
                              void* d_out, int out_size, void* d_ws, size_t ws_size,
                              hipStream_t stream) {
    const float* states  = (const float*)d_in[0];   // [32, 512, 2048]
    const float* actions = (const float*)d_in[1];   // [16, 512, 2048]
    const float* rewards = (const float*)d_in[2];   // [1, 512, 2048]
    const float* W       = (const float*)d_in[3];   // [16, 32]
    const float* log_std = (const float*)d_in[4];   // [16]

    float* out = (float*)d_out;                     // [0]=loss, [1..512]=expected_reward

    char* ws = (char*)d_ws;
    float* lossAcc = (float*)(ws + WS_LOSS_OFF);
    float* csum    = (float*)(ws + WS_CSUM_OFF);
    float* tail    = (float*)(ws + WS_TAIL_OFF);
    float* rowsum  = (float*)(ws + WS_ROWSUM_OFF);

    pg_init<<<1, 1, 0, stream>>>(lossAcc);

    pg_chunk_sum<<<(NCHUNK * S_LEN) / 256, 256, 0, stream>>>(rewards, csum);
    pg_tail<<<S_LEN / 256, 256, 0, stream>>>(csum, tail);

    pg_row_sum<<<T_LEN, 256, 0, stream>>>(rewards, rowsum);
    pg_expected<<<1, 1, 0, stream>>>(rowsum, out + 1);

    dim3 grid(S_LEN / 128, NCHUNK);   // 16 x 16 blocks, 8 waves each
    pg_main<<<grid, 256, 0, stream>>>(states, actions, rewards, W, log_std,
                                      tail, lossAcc);

    pg_finalize<<<1, 1, 0, stream>>>(lossAcc, out);
}